// SimpleTemporalGNN_88373247083010
// MI455X (gfx1250) — compile-verified
//
#include <hip/hip_runtime.h>
#include <hip/hip_bf16.h>
#include <math.h>

#define Bdim 16
#define Sdim 64
#define Ndim 128
#define Fdim 16
#define Hdim 64
#define Edim 1024
#define Ldim 2
#define EtDim (Edim + Ndim)   // 1152 edges incl. self loops
#define NEG_SLOPE 0.2f
#define KPAD 4                // LDS row padding (floats) -> conflict-free B reads

typedef float v2f __attribute__((ext_vector_type(2)));
typedef float v8f __attribute__((ext_vector_type(8)));
typedef int   v4i __attribute__((ext_vector_type(4)));

// --------- gfx1250 async Global->LDS staging (guarded; fallback = plain copy) ----------
#if defined(__has_builtin)
#  if __has_builtin(__builtin_amdgcn_global_load_async_to_lds_b128)
#    define HAVE_ASYNC_LDS 1
#  endif
#endif
#ifndef HAVE_ASYNC_LDS
#  define HAVE_ASYNC_LDS 0
#endif

__device__ __forceinline__ void wait_asynccnt0()
{
#if defined(__has_builtin)
#  if __has_builtin(__builtin_amdgcn_s_wait_asynccnt)
  __builtin_amdgcn_s_wait_asynccnt(0);
#  else
  asm volatile("s_wait_asynccnt 0x0" ::: "memory");
#  endif
#else
  asm volatile("s_wait_asynccnt 0x0" ::: "memory");
#endif
}

// ---------------------------------------------------------------------------
// fp32 WMMA GEMM:  C[M,Nc] = A[M,K] (row stride lda) * W[Nc,K]^T + bias
// Block = 256 threads (8 waves); each wave owns NT 16x16 accumulator tiles and
// reuses one A fragment for NT WMMAs per K-step (V_WMMA_F32_16X16X4_F32).
// W tile staged once per block into LDS via async DMA (ASYNCcnt) with row
// stride K+KPAD floats so B-fragment ds_loads are bank-conflict-free
// (bank = 4*l16 + ka mod 64 -> 64 distinct banks across the wave).
// Layouts per CDNA5 ISA 7.12.2:
//   A 16x4 : lanes 0-15 rows; VGPR0 = K{k0 | k0+2}, VGPR1 = K{k0+1 | k0+3}
//   B 4x16 : lanes 0-15 cols; same K split across lane halves
//   C/D    : 8 VGPRs; lane<16 -> M=r, lane>=16 -> M=r+8; N = lane%16
// ---------------------------------------------------------------------------
template <int NT>
__global__ void gemm_wmma(const float* __restrict__ A, long lda,
                          const float* __restrict__ W,
                          const float* __restrict__ bias,
                          float* __restrict__ C,
                          int Nc, int K, int act)
{
  __shared__ float sW[4 * 16 * (64 + KPAD)];   // up to NT=4, K=64 (17 KB)
  const int tid  = threadIdx.x;
  const int wave = tid >> 5;
  const int lane = tid & 31;
  const int half = lane >> 4;
  const int l16  = lane & 15;
  const int colBase = blockIdx.y * (NT * 16);
  const int ldb  = K + KPAD;                   // padded LDS row stride (floats)

  // Stage W rows [colBase, colBase+NT*16) into LDS, 4-float chunks per thread.
  {
    const float* src = W + (long)colBase * K;
    const int kchunks = K >> 2;                // 4 or 16
    const int nChunk  = NT * 16 * kchunks;     // 256/512/1024 -> uniform loop
    for (int c = tid; c < nChunk; c += 256) {
      const int row = c / kchunks;
      const int ch  = (c - row * kchunks) << 2;
#if HAVE_ASYNC_LDS
      __builtin_amdgcn_global_load_async_to_lds_b128(
          (__attribute__((address_space(1))) v4i*)(src + (long)row * K + ch),
          (__attribute__((address_space(3))) v4i*)(sW + row * ldb + ch), 0, 0);
#else
      const float* s4 = src + (long)row * K + ch;
      float*       d4 = sW + row * ldb + ch;
      d4[0] = s4[0]; d4[1] = s4[1]; d4[2] = s4[2]; d4[3] = s4[3];
#endif
    }
#if HAVE_ASYNC_LDS
    wait_asynccnt0();
#endif
    __syncthreads();
  }

  const long mt   = (long)blockIdx.x * 8 + wave;
  const long arow = mt * 16 + l16;

  v8f acc[NT];
  const v8f vzero = {0.f, 0.f, 0.f, 0.f, 0.f, 0.f, 0.f, 0.f};
#pragma unroll
  for (int t = 0; t < NT; ++t) acc[t] = vzero;

  for (int k0 = 0; k0 < K; k0 += 4) {
    const int ka = k0 + half * 2;
    v2f a;
    a.x = A[arow * lda + ka];
    a.y = A[arow * lda + ka + 1];
    v2f b[NT];
#pragma unroll
    for (int t = 0; t < NT; ++t) {
      b[t].x = sW[(t * 16 + l16) * ldb + ka];
      b[t].y = sW[(t * 16 + l16) * ldb + ka + 1];
    }
#pragma unroll
    for (int t = 0; t < NT; ++t)
      acc[t] = __builtin_amdgcn_wmma_f32_16x16x4_f32(
          false, a, false, b[t], (short)0, acc[t], false, false);
  }

#pragma unroll
  for (int t = 0; t < NT; ++t) {
    const int col = colBase + t * 16 + l16;
    const float bb = bias ? bias[col] : 0.0f;
#pragma unroll
    for (int r = 0; r < 8; ++r) {
      const long row = mt * 16 + r + half * 8;
      float v = acc[t][r] + bb;
      if (act) v = fmaxf(v, 0.0f);
      C[row * (long)Nc + col] = v;
    }
  }
}

// Final head layer (Nc == 1): out[r] = b + X[r,:] . w
__global__ void head_dot(const float* __restrict__ X, const float* __restrict__ w,
                         const float* __restrict__ b, float* __restrict__ out, int K)
{
  const int r = blockIdx.x * blockDim.x + threadIdx.x;
  if (r >= Bdim * Ndim) return;
  float acc = b[0];
  for (int k = 0; k < K; ++k) acc += X[(long)r * K + k] * w[k];
  out[r] = acc;
}

// ---------------------------------------------------------------------------
// Build edge arrays with self-loops + deterministic CSR grouped by target.
// ---------------------------------------------------------------------------
__global__ void build_csr(const int* __restrict__ edge_index,
                          int* __restrict__ srcA, int* __restrict__ tgtA,
                          int* __restrict__ starts, int* __restrict__ eids)
{
  __shared__ int cnt[Ndim];
  __shared__ int stg[EtDim];
  const int tid = threadIdx.x;
  for (int i = tid; i < Ndim; i += blockDim.x) cnt[i] = 0;
  __syncthreads();
  for (int e = tid; e < EtDim; e += blockDim.x) {
    const int s = (e < Edim) ? edge_index[e]        : (e - Edim);
    const int t = (e < Edim) ? edge_index[Edim + e] : (e - Edim);
    srcA[e] = s; tgtA[e] = t; stg[e] = t;
    atomicAdd(&cnt[t], 1);
  }
  __syncthreads();
  if (tid == 0) {
    int run = 0;
    for (int i = 0; i < Ndim; ++i) { starts[i] = run; run += cnt[i]; }
    starts[Ndim] = run;
  }
  __syncthreads();
  for (int e = tid; e < EtDim; e += blockDim.x) {
    const int t = stg[e];
    int rank = 0;
    for (int e2 = 0; e2 < e; ++e2) rank += (stg[e2] == t) ? 1 : 0;
    eids[starts[t] + rank] = e;
  }
}

// ---------------------------------------------------------------------------
// Attention logits: a[g,e] = att . leaky_relu(xl[g,src] + xr[g,tgt])
// ---------------------------------------------------------------------------
__global__ void gat_scores(const float* __restrict__ xl, const float* __restrict__ xr,
                           const float* __restrict__ att,
                           const int* __restrict__ srcA, const int* __restrict__ tgtA,
                           float* __restrict__ scores)
{
  __shared__ float satt[Hdim];
  const int  g    = blockIdx.x;
  const long base = (long)g * Ndim * Hdim;
  for (int i = threadIdx.x; i < Hdim; i += blockDim.x) satt[i] = att[i];
  __syncthreads();
  for (int e = threadIdx.x; e < EtDim; e += blockDim.x) {
    const float* pl = xl + base + (long)srcA[e] * Hdim;
    const float* pr = xr + base + (long)tgtA[e] * Hdim;
    float acc = 0.0f;
    for (int h = 0; h < Hdim; ++h) {
      float v = pl[h] + pr[h];
      v = (v > 0.0f) ? v : NEG_SLOPE * v;
      acc += satt[h] * v;
    }
    scores[(long)g * EtDim + e] = acc;
  }
}

// ---------------------------------------------------------------------------
// Segment softmax over targets + weighted aggregation of xl[src] + bias + relu.
// One wave per target; lane owns 2 channels of H=64.
// ---------------------------------------------------------------------------
__global__ void gat_aggregate(const float* __restrict__ xl,
                              const float* __restrict__ scores,
                              const int* __restrict__ srcA,
                              const int* __restrict__ starts,
                              const int* __restrict__ eids,
                              const float* __restrict__ bo,
                              float* __restrict__ hout)
{
  const int  g    = blockIdx.x;
  const long base = (long)g * Ndim * Hdim;
  const float* sc = scores + (long)g * EtDim;
  const int wave   = threadIdx.x >> 5;
  const int lane   = threadIdx.x & 31;
  const int nwaves = blockDim.x >> 5;
  const int c0 = lane * 2;
  for (int t = wave; t < Ndim; t += nwaves) {
    const int jb = starts[t], je = starts[t + 1];
    float m = -INFINITY;
    for (int j = jb; j < je; ++j) m = fmaxf(m, sc[eids[j]]);
    float z = 0.0f, a0 = 0.0f, a1 = 0.0f;
    for (int j = jb; j < je; ++j) {
      const int e = eids[j];
      const float p = expf(sc[e] - m);
      z += p;
      const float* ps = xl + base + (long)srcA[e] * Hdim;
      a0 += p * ps[c0];
      a1 += p * ps[c0 + 1];
    }
    const float inv = 1.0f / z;
    const float o0 = a0 * inv + bo[c0];
    const float o1 = a1 * inv + bo[c0 + 1];
    hout[base + (long)t * Hdim + c0]     = fmaxf(o0, 0.0f);
    hout[base + (long)t * Hdim + c0 + 1] = fmaxf(o1, 0.0f);
  }
}

// [B,S,N,H] -> [B*N, S, H]
__global__ void transpose_sn(const float* __restrict__ hin, float* __restrict__ seqs)
{
  const long total = (long)Bdim * Sdim * Ndim * Hdim;
  long i = (long)blockIdx.x * blockDim.x + threadIdx.x;
  if (i >= total) return;
  const int h = (int)(i & (Hdim - 1));
  long rest = i >> 6;
  const int n = (int)(rest & (Ndim - 1));
  rest >>= 7;
  const int t = (int)(rest & (Sdim - 1));
  const int b = (int)(rest >> 6);
  seqs[(((long)(b * Ndim + n) * Sdim) + t) * Hdim + h] = hin[i];
}

__global__ void zero_f32(float* __restrict__ p, long n)
{
  const long i = (long)blockIdx.x * blockDim.x + threadIdx.x;
  if (i < n) p[i] = 0.0f;
}

// Fused GRU gate math; updates H in place (each thread owns one element).
__global__ void gru_gate(const float* __restrict__ GI, const float* __restrict__ GH,
                         float* __restrict__ Hst)
{
  const int R = Bdim * Ndim;
  const int i = blockIdx.x * blockDim.x + threadIdx.x;
  if (i >= R * Hdim) return;
  const int r = i >> 6, c = i & 63;
  const long rb = (long)r * 192;
  const float gir = GI[rb + c], giz = GI[rb + 64 + c], gin = GI[rb + 128 + c];
  const float ghr = GH[rb + c], ghz = GH[rb + 64 + c], ghn = GH[rb + 128 + c];
  const float rr = 1.0f / (1.0f + expf(-(gir + ghr)));
  const float zz = 1.0f / (1.0f + expf(-(giz + ghz)));
  const float nn = tanhf(gin + rr * ghn);
  Hst[i] = (1.0f - zz) * nn + zz * Hst[i];
}

// ---------------------------------------------------------------------------
extern "C" void kernel_launch(void* const* d_in, const int* in_sizes, int n_in,
                              void* d_out, int out_size, void* d_ws, size_t ws_size,
                              hipStream_t stream)
{
  const float* x        = (const float*)d_in[0];
  const int*   eidx     = (const int*)  d_in[1];
  const float* in_W     = (const float*)d_in[2];
  const float* in_b     = (const float*)d_in[3];
  const float* gat_Wl   = (const float*)d_in[4];
  const float* gat_bl   = (const float*)d_in[5];
  const float* gat_Wr   = (const float*)d_in[6];
  const float* gat_br   = (const float*)d_in[7];
  const float* gat_att  = (const float*)d_in[8];
  const float* gat_bias = (const float*)d_in[9];
  const float* gru_Wih  = (const float*)d_in[10];
  const float* gru_Whh  = (const float*)d_in[11];
  const float* gru_bih  = (const float*)d_in[12];
  const float* gru_bhh  = (const float*)d_in[13];
  const float* oh_W1    = (const float*)d_in[14];
  const float* oh_b1    = (const float*)d_in[15];
  const float* oh_W2    = (const float*)d_in[16];
  const float* oh_b2    = (const float*)d_in[17];
  const float* dh_W1    = (const float*)d_in[18];
  const float* dh_b1    = (const float*)d_in[19];
  const float* dh_W2    = (const float*)d_in[20];
  const float* dh_b2    = (const float*)d_in[21];

  const long MBSN = (long)Bdim * Sdim * Ndim;            // 131072 rows
  const long featBytes = MBSN * Hdim * sizeof(float);    // 33.5 MB

  char*  ws    = (char*)d_ws;
  float* hbuf  = (float*)(ws);                 // current node features
  float* xlbuf = (float*)(ws + featBytes);     // xl; later reused as seqs
  float* xrbuf = (float*)(ws + 2 * featBytes); // xr; later reused for GRU scratch
  char*  misc  = ws + 3 * featBytes;

  float* scores = (float*)misc;                                  // [B*S, Et]
  int*   srcA   = (int*)(misc + (long)Bdim * Sdim * EtDim * 4);
  int*   tgtA   = srcA + EtDim;
  int*   starts = tgtA + EtDim;                                  // Ndim+1 ints
  int*   eids   = starts + (Ndim + 4);

  const int R = Bdim * Ndim;                   // 2048 GRU rows
  float* seqs = xlbuf;
  float* GI   = xrbuf;                         // [R,192]
  float* GH   = GI + (long)R * 192;            // [R,192]
  float* Hst  = GH + (long)R * 192;            // [R,64]
  float* T1   = Hst + (long)R * Hdim;          // [R,32]
  float* outF = (float*)d_out;

  build_csr<<<1, 256, 0, stream>>>(eidx, srcA, tgtA, starts, eids);

  // input projection: h = x @ in_W^T + in_b    [131072,16] x [16,64]
  gemm_wmma<4><<<dim3((unsigned)(MBSN / 128), 1), 256, 0, stream>>>(
      x, Fdim, in_W, in_b, hbuf, Hdim, Fdim, 0);

  for (int l = 0; l < Ldim; ++l) {
    const float* Wl = gat_Wl + (long)l * Hdim * Hdim;
    const float* bl = gat_bl + (long)l * Hdim;
    const float* Wr = gat_Wr + (long)l * Hdim * Hdim;
    const float* br = gat_br + (long)l * Hdim;
    gemm_wmma<4><<<dim3((unsigned)(MBSN / 128), 1), 256, 0, stream>>>(
        hbuf, Hdim, Wl, bl, xlbuf, Hdim, Hdim, 0);
    gemm_wmma<4><<<dim3((unsigned)(MBSN / 128), 1), 256, 0, stream>>>(
        hbuf, Hdim, Wr, br, xrbuf, Hdim, Hdim, 0);
    gat_scores<<<Bdim * Sdim, 256, 0, stream>>>(
        xlbuf, xrbuf, gat_att + (long)l * Hdim, srcA, tgtA, scores);
    gat_aggregate<<<Bdim * Sdim, 256, 0, stream>>>(
        xlbuf, scores, srcA, starts, eids, gat_bias + (long)l * Hdim, hbuf);
  }

  // [B,S,N,H] -> [B*N,S,H]
  {
    const long total = MBSN * Hdim;
    transpose_sn<<<(unsigned)((total + 255) / 256), 256, 0, stream>>>(hbuf, seqs);
  }

  // GRU over S steps; x_t view = seqs + t*H with row stride S*H (no gather copy)
  zero_f32<<<(R * Hdim + 255) / 256, 256, 0, stream>>>(Hst, (long)R * Hdim);
  for (int t = 0; t < Sdim; ++t) {
    gemm_wmma<4><<<dim3(R / 128, 3), 256, 0, stream>>>(
        seqs + (long)t * Hdim, (long)Sdim * Hdim, gru_Wih, gru_bih, GI, 192, Hdim, 0);
    gemm_wmma<4><<<dim3(R / 128, 3), 256, 0, stream>>>(
        Hst, Hdim, gru_Whh, gru_bhh, GH, 192, Hdim, 0);
    gru_gate<<<(R * Hdim + 255) / 256, 256, 0, stream>>>(GI, GH, Hst);
  }

  // heads: order then demand into d_out (each [B*N] = 2048)
  gemm_wmma<2><<<dim3(R / 128, 1), 256, 0, stream>>>(Hst, Hdim, oh_W1, oh_b1, T1, 32, Hdim, 1);
  head_dot<<<(R + 255) / 256, 256, 0, stream>>>(T1, oh_W2, oh_b2, outF, 32);
  gemm_wmma<2><<<dim3(R / 128, 1), 256, 0, stream>>>(Hst, Hdim, dh_W1, dh_b1, T1, 32, Hdim, 1);
  head_dot<<<(R + 255) / 256, 256, 0, stream>>>(T1, dh_W2, dh_b2, outF + R, 32);
}